// AI4CFD_90778428768640
// MI455X (gfx1250) — compile-verified
//
#include <hip/hip_runtime.h>

// ---------------------------------------------------------------------------
// AI4CFD NSE step + multigrid F-cycle for MI455X (gfx1250, wave32).
// fp32 stencil workload -> bandwidth bound (23.3 TB/s HBM, working set ~L2).
// Multigrid core (residual / Jacobi / smooth+prolong) runs its plus-shaped
// 3x3 stencils on the matrix pipe: out_tile = SY*T + T*SX via
// V_WMMA_F32_16X16X4_F32 (8 chained K=4 WMMAs per 16x16 tile) + VALU halo fix.
// Tile fills use GLOBAL_LOAD_ASYNC_TO_LDS_B32 (ASYNCcnt) when available:
// global->LDS DMA without touching VGPRs, overlapped with other waves' WMMA.
// ---------------------------------------------------------------------------

#define NN   2048
#define DT   0.01f
#define RE   1.0f

typedef float v2f __attribute__((ext_vector_type(2)));
typedef float v8f __attribute__((ext_vector_type(8)));

#if defined(__HIP_DEVICE_COMPILE__) && defined(__gfx1250__) && \
    __has_builtin(__builtin_amdgcn_global_load_async_to_lds_b32)
#define HAVE_ASYNC_LDS 1
typedef __attribute__((address_space(1))) int* as1p;   // builtin takes AS1 int*
typedef __attribute__((address_space(3))) int* as3p;   // builtin takes AS3 int*
#else
#define HAVE_ASYNC_LDS 0
#endif

// Copy one f32 global->LDS; invalid -> write 0 to LDS (pad_zero semantics).
__device__ __forceinline__ void g2l_f32(const float* __restrict__ gsrc,
                                        float* ldst, bool valid) {
#if HAVE_ASYNC_LDS
    if (valid) {
        __builtin_amdgcn_global_load_async_to_lds_b32((as1p)gsrc, (as3p)ldst, 0, 0);
    } else {
        *ldst = 0.f;
    }
#else
    *ldst = valid ? *gsrc : 0.f;
#endif
}

__device__ __forceinline__ void wait_async_lds() {
#if HAVE_ASYNC_LDS
    asm volatile("s_wait_asynccnt 0" ::: "memory");
#endif
}

#define LDSP 34   // 32x32 block tile + halo 1

// Fill a 34x34 halo tile in LDS. edge=true -> clamp (pad_edge), else zero pad.
__device__ __forceinline__ void load_tile34(const float* __restrict__ src, int n,
                                            int oy, int ox, float* lds, bool edge) {
    for (int idx = threadIdx.x; idx < LDSP * LDSP; idx += blockDim.x) {
        int r = idx / LDSP, c = idx - r * LDSP;
        int gy = oy + r - 1, gx = ox + c - 1;
        if (edge) {
            int cy = gy < 0 ? 0 : (gy > n - 1 ? n - 1 : gy);
            int cx = gx < 0 ? 0 : (gx > n - 1 ? n - 1 : gx);
            g2l_f32(&src[(size_t)cy * n + cx], &lds[idx], true);
        } else {
            bool in = (gy >= 0 && gy < n && gx >= 0 && gx < n);
            g2l_f32(&src[in ? ((size_t)gy * n + gx) : 0], &lds[idx], in);
        }
    }
}

// Plus-shaped 3x3 stencil on one 16x16 tile via f32 WMMA.
// OUT = SY*T + T*SX :  SX[n-1,n]=wl, SX[n,n]=wc, SX[n+1,n]=wr ;
//                      SY[m,m-1]=wu, SY[m,m+1]=wd.
// T read from the shared 34x34 LDS tile at wave-tile origin (ty,tx).
// Halo contributions (cols 0/15, rows 0/15) fixed up with VALU from LDS halo.
__device__ __forceinline__ v8f wmma_plus_stencil(const float* lds, int ty, int tx,
                                                 float wu, float wl, float wc,
                                                 float wr, float wd) {
    const int lane = threadIdx.x & 31;
    const int lm = lane & 15;    // column (B/C/D) or row (A)
    const int lh = lane >> 4;    // half-wave select
    v8f acc = {0.f, 0.f, 0.f, 0.f, 0.f, 0.f, 0.f, 0.f};
#pragma unroll
    for (int q = 0; q < 4; ++q) {          // K in chunks of 4
        v2f aT, bT, aS, bS;
#pragma unroll
        for (int e = 0; e < 2; ++e) {
            int k = 4 * q + e + 2 * lh;    // per ISA A/B f32 VGPR layout
            // T in A layout: T[lm][k]
            aT[e] = lds[(ty + lm + 1) * LDSP + (tx + k + 1)];
            // T in B layout: T[k][lm]
            bT[e] = lds[(ty + k + 1) * LDSP + (tx + lm + 1)];
            // SX in B layout: SX[k][lm]
            float sx = 0.f;
            sx = (k == lm - 1) ? wl : sx;
            sx = (k == lm)     ? wc : sx;
            sx = (k == lm + 1) ? wr : sx;
            bS[e] = sx;
            // SY in A layout: SY[lm][k]
            float sy = 0.f;
            sy = (k == lm - 1) ? wu : sy;
            sy = (k == lm + 1) ? wd : sy;
            aS[e] = sy;
        }
        acc = __builtin_amdgcn_wmma_f32_16x16x4_f32(false, aT, false, bS,
                                                    (short)0, acc, false, false);
        acc = __builtin_amdgcn_wmma_f32_16x16x4_f32(false, aS, false, bT,
                                                    (short)0, acc, false, false);
    }
    // halo fixups; C/D element e: row = e + 8*lh, col = lm
#pragma unroll
    for (int e = 0; e < 8; ++e) {
        int row = e + 8 * lh;
        float fix = 0.f;
        if (lm == 0)   fix += wl * lds[(ty + row + 1) * LDSP + (tx + 0)];
        if (lm == 15)  fix += wr * lds[(ty + row + 1) * LDSP + (tx + 17)];
        if (row == 0)  fix += wu * lds[(ty + 0) * LDSP + (tx + lm + 1)];
        if (row == 15) fix += wd * lds[(ty + 17) * LDSP + (tx + lm + 1)];
        acc[e] += fix;
    }
    return acc;
}

// r = conv(pad_edge(p), w_A) - b          (2048^2, WMMA path)
__global__ __launch_bounds__(128) void residual_wmma_k(
        const float* __restrict__ p, const float* __restrict__ b,
        const float* __restrict__ WA, int n, float* __restrict__ r) {
    __shared__ float lds[LDSP * LDSP];
    const int ox = blockIdx.x * 32, oy = blockIdx.y * 32;
    load_tile34(p, n, oy, ox, lds, true);
    wait_async_lds();
    __syncthreads();
    const int wave = threadIdx.x >> 5;
    const int ty = (wave >> 1) * 16, tx = (wave & 1) * 16;
    v8f acc = wmma_plus_stencil(lds, ty, tx, WA[1], WA[3], WA[4], WA[5], WA[7]);
    const int lane = threadIdx.x & 31, lm = lane & 15, lh = lane >> 4;
#pragma unroll
    for (int e = 0; e < 8; ++e) {
        int gy = oy + ty + e + 8 * lh;
        int gx = ox + tx + lm;
        size_t g = (size_t)gy * n + gx;
        r[g] = acc[e] - b[g];
    }
}

// p_dst = p - conv(pad_edge(p), w_A)/diag + b/diag   (2048^2, WMMA path)
__global__ __launch_bounds__(128) void jacobi_wmma_k(
        const float* __restrict__ ps, const float* __restrict__ b,
        const float* __restrict__ WA, int n, float* __restrict__ pd) {
    __shared__ float lds[LDSP * LDSP];
    const int ox = blockIdx.x * 32, oy = blockIdx.y * 32;
    load_tile34(ps, n, oy, ox, lds, true);
    wait_async_lds();
    __syncthreads();
    const int wave = threadIdx.x >> 5;
    const int ty = (wave >> 1) * 16, tx = (wave & 1) * 16;
    const float diag = WA[4];
    v8f acc = wmma_plus_stencil(lds, ty, tx, WA[1], WA[3], WA[4], WA[5], WA[7]);
    const int lane = threadIdx.x & 31, lm = lane & 15, lh = lane >> 4;
#pragma unroll
    for (int e = 0; e < 8; ++e) {
        int row = e + 8 * lh;
        int gy = oy + ty + row, gx = ox + tx + lm;
        size_t g = (size_t)gy * n + gx;
        float pc = lds[(ty + row + 1) * LDSP + (tx + lm + 1)];
        pd[g] = pc - acc[e] / diag + b[g] / diag;
    }
}

// wdst(2s x 2s) = upsample2( w - conv(pad_zero(w), w_A)/diag + r/diag ), s>=32
__global__ __launch_bounds__(128) void smooth_up_wmma_k(
        const float* __restrict__ wsrc, const float* __restrict__ rl,
        const float* __restrict__ WA, int s, float* __restrict__ wdst) {
    __shared__ float lds[LDSP * LDSP];
    const int ox = blockIdx.x * 32, oy = blockIdx.y * 32;
    load_tile34(wsrc, s, oy, ox, lds, false);   // pad_zero
    wait_async_lds();
    __syncthreads();
    const int wave = threadIdx.x >> 5;
    const int ty = (wave >> 1) * 16, tx = (wave & 1) * 16;
    const float diag = WA[4];
    v8f acc = wmma_plus_stencil(lds, ty, tx, WA[1], WA[3], WA[4], WA[5], WA[7]);
    const int lane = threadIdx.x & 31, lm = lane & 15, lh = lane >> 4;
    const int o = 2 * s;
#pragma unroll
    for (int e = 0; e < 8; ++e) {
        int row = e + 8 * lh;
        int gy = oy + ty + row, gx = ox + tx + lm;
        float wcv = lds[(ty + row + 1) * LDSP + (tx + lm + 1)];
        float val = wcv - acc[e] / diag + rl[(size_t)gy * s + gx] / diag;
        size_t base = (size_t)(2 * gy) * o + 2 * gx;
        wdst[base] = val; wdst[base + 1] = val;
        wdst[base + o] = val; wdst[base + o + 1] = val;
    }
}

// Same smoother for tiny levels (s = 2..16), plain VALU, one block of s*s threads
__global__ void smooth_small_k(const float* __restrict__ wsrc, const float* __restrict__ rl,
                               const float* __restrict__ WA, int s, int wzero,
                               float* __restrict__ wdst) {
    int x = threadIdx.x, y = threadIdx.y;
    if (x >= s || y >= s) return;
    float diag = WA[4];
    float wcv = 0.f, conv = 0.f;
    if (!wzero) {
        wcv = wsrc[y * s + x];
        for (int a = 0; a < 3; ++a)
            for (int c = 0; c < 3; ++c) {
                int gy = y + a - 1, gx = x + c - 1;
                if (gy >= 0 && gy < s && gx >= 0 && gx < s)
                    conv += WA[a * 3 + c] * wsrc[gy * s + gx];
            }
    }
    float val = wcv - conv / diag + rl[y * s + x] / diag;
    int o = 2 * s;
    int base = (2 * y) * o + 2 * x;
    wdst[base] = val; wdst[base + 1] = val;
    wdst[base + o] = val; wdst[base + o + 1] = val;
}

// dst(n/2) = 2x2 weighted restriction of src(n)
__global__ void restrict_k(const float* __restrict__ src, int ni,
                           const float* __restrict__ Wr, float* __restrict__ dst) {
    int gx = blockIdx.x * blockDim.x + threadIdx.x;
    int gy = blockIdx.y * blockDim.y + threadIdx.y;
    int no = ni >> 1;
    if (gx >= no || gy >= no) return;
    size_t b0 = (size_t)(2 * gy) * ni + 2 * gx;
    dst[(size_t)gy * no + gx] =
        Wr[0] * src[b0] + Wr[1] * src[b0 + 1] + Wr[2] * src[b0 + ni] + Wr[3] * src[b0 + ni + 1];
}

// b = -(conv(pad_zero(u), w_xadv) + conv(pad_zero(v), w_yadv)) / DT
__global__ void divb_k(const float* __restrict__ u, const float* __restrict__ v,
                       const float* __restrict__ Wx, const float* __restrict__ Wy,
                       int n, float* __restrict__ b) {
    int gx = blockIdx.x * 16 + threadIdx.x;
    int gy = blockIdx.y * 16 + threadIdx.y;
    if (gx >= n || gy >= n) return;
    float cu = 0.f, cv = 0.f;
    for (int a = 0; a < 3; ++a)
        for (int c = 0; c < 3; ++c) {
            int yy = gy + a - 1, xx = gx + c - 1;
            if (yy >= 0 && yy < n && xx >= 0 && xx < n) {
                size_t g = (size_t)yy * n + xx;
                cu += Wx[a * 3 + c] * u[g];
                cv += Wy[a * 3 + c] * v[g];
            }
        }
    b[(size_t)gy * n + gx] = -(cu + cv) / DT;
}

__global__ void sub_k(const float* __restrict__ a, const float* __restrict__ c,
                      float* __restrict__ d, int n) {
    int i = blockIdx.x * blockDim.x + threadIdx.x;
    if (i < n) d[i] = a[i] - c[i];
}

__global__ void copy_k(const float* __restrict__ a, float* __restrict__ d, int n) {
    int i = blockIdx.x * blockDim.x + threadIdx.x;
    if (i < n) d[i] = a[i];
}

// u -= conv(pad_edge(p), w_xadv)*DT ; v -= conv(pad_edge(p), w_yadv)*DT
__global__ void pgrad_correct_k(const float* __restrict__ p, const float* __restrict__ Wx,
                                const float* __restrict__ Wy, int n,
                                float* __restrict__ u, float* __restrict__ v) {
    int gx = blockIdx.x * 16 + threadIdx.x;
    int gy = blockIdx.y * 16 + threadIdx.y;
    if (gx >= n || gy >= n) return;
    float cx = 0.f, cy = 0.f;
    for (int a = 0; a < 3; ++a)
        for (int c = 0; c < 3; ++c) {
            int yy = gy + a - 1, xx = gx + c - 1;
            yy = yy < 0 ? 0 : (yy > n - 1 ? n - 1 : yy);
            xx = xx < 0 ? 0 : (xx > n - 1 ? n - 1 : xx);
            float pv = p[(size_t)yy * n + xx];
            cx += Wx[a * 3 + c] * pv;
            cy += Wy[a * 3 + c] * pv;
        }
    size_t g0 = (size_t)gy * n + gx;
    u[g0] -= cx * DT;
    v[g0] -= cy * DT;
}

// ------------------- fused momentum pass 1: b_u, b_v -----------------------
// PG limiter needs k at halo-1, which needs velocity stencils at halo-2:
// single kernel with a (16+4)^2 velocity tile, k computed on (16+2)^2 in LDS.
__global__ __launch_bounds__(256) void momentum1_k(
        const float* __restrict__ u, const float* __restrict__ v,
        const float* __restrict__ p, const float* __restrict__ k1,
        const float* __restrict__ Wx, const float* __restrict__ Wy,
        const float* __restrict__ Wd, int n,
        float* __restrict__ bu, float* __restrict__ bv) {
    __shared__ float su[20 * 20], sv[20 * 20];
    __shared__ float sp[18 * 18], sk1[18 * 18], sku[18 * 18], skv[18 * 18];
    const int ox = blockIdx.x * 16, oy = blockIdx.y * 16;
    const int tid = threadIdx.y * 16 + threadIdx.x;
    float wxr[9], wyr[9], wdr[9];
#pragma unroll
    for (int i = 0; i < 9; ++i) { wxr[i] = Wx[i]; wyr[i] = Wy[i]; wdr[i] = Wd[i]; }

    for (int idx = tid; idx < 400; idx += 256) {        // u,v halo-2, pad_zero
        int r = idx / 20, c = idx - r * 20;
        int gy = oy + r - 2, gx = ox + c - 2;
        bool in = (gy >= 0 && gy < n && gx >= 0 && gx < n);
        size_t g = in ? ((size_t)gy * n + gx) : 0;
        g2l_f32(&u[g], &su[idx], in);
        g2l_f32(&v[g], &sv[idx], in);
    }
    for (int idx = tid; idx < 324; idx += 256) {        // p halo-1 edge, k1 clamped
        int r = idx / 18, c = idx - r * 18;
        int gy = oy + r - 1, gx = ox + c - 1;
        int cy = gy < 0 ? 0 : (gy > n - 1 ? n - 1 : gy);
        int cx = gx < 0 ? 0 : (gx > n - 1 ? n - 1 : gx);
        size_t g = (size_t)cy * n + cx;
        g2l_f32(&p[g], &sp[idx], true);
        g2l_f32(&k1[g], &sk1[idx], true);
    }
    wait_async_lds();
    __syncthreads();

    for (int idx = tid; idx < 324; idx += 256) {        // k_u, k_v on halo-1 ring
        int r = idx / 18, c = idx - r * 18;
        int gy = oy + r - 1, gx = ox + c - 1;
        float ku = 0.f, kv = 0.f;
        if (gy >= 0 && gy < n && gx >= 0 && gx < n) {
            float adxu = 0, adyu = 0, ad2u = 0, adxv = 0, adyv = 0, ad2v = 0;
#pragma unroll
            for (int a = 0; a < 3; ++a)
#pragma unroll
                for (int cc = 0; cc < 3; ++cc) {
                    float uu = su[(r + a) * 20 + (c + cc)];
                    float vv = sv[(r + a) * 20 + (c + cc)];
                    int wi = a * 3 + cc;
                    adxu += wxr[wi] * uu; adyu += wyr[wi] * uu; ad2u += wdr[wi] * uu;
                    adxv += wxr[wi] * vv; adyv += wyr[wi] * vv; ad2v += wdr[wi] * vv;
                }
            float uc = su[(r + 1) * 20 + (c + 1)];
            float vc = sv[(r + 1) * 20 + (c + 1)];
            float mag = fabsf(uc) + fabsf(vc);        // DX = DY = 1
            float k1c = sk1[idx];
            ku = 0.25f * fabsf(0.5f * mag * ad2u) /
                 (0.001f + 0.5f * (fabsf(adxu) + fabsf(adyu)));
            kv = 0.25f * fabsf(0.5f * mag * ad2v) /
                 (0.001f + 0.5f * (fabsf(adxv) + fabsf(adyv)));
            ku = fminf(ku, k1c); kv = fminf(kv, k1c);
        }
        sku[idx] = ku; skv[idx] = kv;
    }
    __syncthreads();

    int lx = threadIdx.x, ly = threadIdx.y;
    int gx = ox + lx, gy = oy + ly;
    if (gx >= n || gy >= n) return;
    float adxu = 0, adyu = 0, ad2u = 0, adxv = 0, adyv = 0, ad2v = 0;
    float cAu = 0, cBu = 0, cAv = 0, cBv = 0, gpx = 0, gpy = 0;
#pragma unroll
    for (int a = 0; a < 3; ++a)
#pragma unroll
        for (int cc = 0; cc < 3; ++cc) {
            float uu = su[(ly + 1 + a) * 20 + (lx + 1 + cc)];
            float vv = sv[(ly + 1 + a) * 20 + (lx + 1 + cc)];
            float kun = sku[(ly + a) * 18 + (lx + cc)];
            float kvn = skv[(ly + a) * 18 + (lx + cc)];
            float pp = sp[(ly + a) * 18 + (lx + cc)];
            int wi = a * 3 + cc;
            adxu += wxr[wi] * uu; adyu += wyr[wi] * uu; ad2u += wdr[wi] * uu;
            adxv += wxr[wi] * vv; adyv += wyr[wi] * vv; ad2v += wdr[wi] * vv;
            cAu += wdr[wi] * uu * kun; cBu += wdr[wi] * kun;
            cAv += wdr[wi] * vv * kvn; cBv += wdr[wi] * kvn;
            gpx += wxr[wi] * pp; gpy += wyr[wi] * pp;
        }
    float uc = su[(ly + 2) * 20 + lx + 2];
    float vc = sv[(ly + 2) * 20 + lx + 2];
    float kuc = sku[(ly + 1) * 18 + lx + 1];
    float kvc = skv[(ly + 1) * 18 + lx + 1];
    float kx = 1.5f * (kuc * ad2u + cAu - uc * cBu);
    float ky = 1.5f * (kvc * ad2v + cAv - vc * cBv);
    size_t g0 = (size_t)gy * n + gx;
    bu[g0] = uc + 0.5f * (RE * kx * DT - uc * adxu * DT - vc * adyu * DT) - gpx * DT;
    bv[g0] = vc + 0.5f * (RE * ky * DT - uc * adxv * DT - vc * adyv * DT) - gpy * DT;
}

// ------------------- fused momentum pass 2: corrector ----------------------
// k from (b_u, b_v); product conv uses ORIGINAL velocities (torch globals).
__global__ __launch_bounds__(256) void momentum2_k(
        const float* __restrict__ u, const float* __restrict__ v,
        const float* __restrict__ p, const float* __restrict__ bu,
        const float* __restrict__ bv, const float* __restrict__ k1,
        const float* __restrict__ Fx, const float* __restrict__ Fy,
        const float* __restrict__ Wx, const float* __restrict__ Wy,
        const float* __restrict__ Wd, int n,
        float* __restrict__ uo, float* __restrict__ vo) {
    __shared__ float sbu[20 * 20], sbv[20 * 20];
    __shared__ float su[18 * 18], sv[18 * 18], sp[18 * 18], sk1[18 * 18];
    __shared__ float sku[18 * 18], skv[18 * 18];
    const int ox = blockIdx.x * 16, oy = blockIdx.y * 16;
    const int tid = threadIdx.y * 16 + threadIdx.x;
    float wxr[9], wyr[9], wdr[9];
#pragma unroll
    for (int i = 0; i < 9; ++i) { wxr[i] = Wx[i]; wyr[i] = Wy[i]; wdr[i] = Wd[i]; }

    for (int idx = tid; idx < 400; idx += 256) {        // b_u, b_v halo-2 pad_zero
        int r = idx / 20, c = idx - r * 20;
        int gy = oy + r - 2, gx = ox + c - 2;
        bool in = (gy >= 0 && gy < n && gx >= 0 && gx < n);
        size_t g = in ? ((size_t)gy * n + gx) : 0;
        g2l_f32(&bu[g], &sbu[idx], in);
        g2l_f32(&bv[g], &sbv[idx], in);
    }
    for (int idx = tid; idx < 324; idx += 256) {        // u,v halo-1 pad_zero; p edge; k1 clamp
        int r = idx / 18, c = idx - r * 18;
        int gy = oy + r - 1, gx = ox + c - 1;
        bool in = (gy >= 0 && gy < n && gx >= 0 && gx < n);
        size_t gz = in ? ((size_t)gy * n + gx) : 0;
        g2l_f32(&u[gz], &su[idx], in);
        g2l_f32(&v[gz], &sv[idx], in);
        int cy = gy < 0 ? 0 : (gy > n - 1 ? n - 1 : gy);
        int cx = gx < 0 ? 0 : (gx > n - 1 ? n - 1 : gx);
        size_t g = (size_t)cy * n + cx;
        g2l_f32(&p[g], &sp[idx], true);
        g2l_f32(&k1[g], &sk1[idx], true);
    }
    wait_async_lds();
    __syncthreads();

    for (int idx = tid; idx < 324; idx += 256) {        // k from b_u, b_v
        int r = idx / 18, c = idx - r * 18;
        int gy = oy + r - 1, gx = ox + c - 1;
        float ku = 0.f, kv = 0.f;
        if (gy >= 0 && gy < n && gx >= 0 && gx < n) {
            float adxu = 0, adyu = 0, ad2u = 0, adxv = 0, adyv = 0, ad2v = 0;
#pragma unroll
            for (int a = 0; a < 3; ++a)
#pragma unroll
                for (int cc = 0; cc < 3; ++cc) {
                    float uu = sbu[(r + a) * 20 + (c + cc)];
                    float vv = sbv[(r + a) * 20 + (c + cc)];
                    int wi = a * 3 + cc;
                    adxu += wxr[wi] * uu; adyu += wyr[wi] * uu; ad2u += wdr[wi] * uu;
                    adxv += wxr[wi] * vv; adyv += wyr[wi] * vv; ad2v += wdr[wi] * vv;
                }
            float uc = sbu[(r + 1) * 20 + (c + 1)];
            float vc = sbv[(r + 1) * 20 + (c + 1)];
            float mag = fabsf(uc) + fabsf(vc);
            float k1c = sk1[idx];
            ku = 0.25f * fabsf(0.5f * mag * ad2u) /
                 (0.001f + 0.5f * (fabsf(adxu) + fabsf(adyu)));
            kv = 0.25f * fabsf(0.5f * mag * ad2v) /
                 (0.001f + 0.5f * (fabsf(adxv) + fabsf(adyv)));
            ku = fminf(ku, k1c); kv = fminf(kv, k1c);
        }
        sku[idx] = ku; skv[idx] = kv;
    }
    __syncthreads();

    int lx = threadIdx.x, ly = threadIdx.y;
    int gx = ox + lx, gy = oy + ly;
    if (gx >= n || gy >= n) return;
    float adxu = 0, adyu = 0, ad2u = 0, adxv = 0, adyv = 0, ad2v = 0;
    float cAu = 0, cBu = 0, cAv = 0, cBv = 0, gpx = 0, gpy = 0;
#pragma unroll
    for (int a = 0; a < 3; ++a)
#pragma unroll
        for (int cc = 0; cc < 3; ++cc) {
            float uu = sbu[(ly + 1 + a) * 20 + (lx + 1 + cc)];
            float vv = sbv[(ly + 1 + a) * 20 + (lx + 1 + cc)];
            float uon = su[(ly + a) * 18 + (lx + cc)];   // original velocities
            float von = sv[(ly + a) * 18 + (lx + cc)];
            float kun = sku[(ly + a) * 18 + (lx + cc)];
            float kvn = skv[(ly + a) * 18 + (lx + cc)];
            float pp = sp[(ly + a) * 18 + (lx + cc)];
            int wi = a * 3 + cc;
            adxu += wxr[wi] * uu; adyu += wyr[wi] * uu; ad2u += wdr[wi] * uu;
            adxv += wxr[wi] * vv; adyv += wyr[wi] * vv; ad2v += wdr[wi] * vv;
            cAu += wdr[wi] * uon * kun; cBu += wdr[wi] * kun;
            cAv += wdr[wi] * von * kvn; cBv += wdr[wi] * kvn;
            gpx += wxr[wi] * pp; gpy += wyr[wi] * pp;
        }
    float buc = sbu[(ly + 2) * 20 + lx + 2];
    float bvc = sbv[(ly + 2) * 20 + lx + 2];
    float ucn = su[(ly + 1) * 18 + lx + 1];
    float vcn = sv[(ly + 1) * 18 + lx + 1];
    float kuc = sku[(ly + 1) * 18 + lx + 1];
    float kvc = skv[(ly + 1) * 18 + lx + 1];
    float kx = 1.5f * (kuc * ad2u + cAu - buc * cBu);
    float ky = 1.5f * (kvc * ad2v + cAv - bvc * cBv);
    size_t g0 = (size_t)gy * n + gx;
    uo[g0] = ucn + RE * kx * DT - buc * adxu * DT - bvc * adyu * DT - gpx * DT - Fx[g0] * DT;
    vo[g0] = vcn + RE * ky * DT - buc * adxv * DT - bvc * adyv * DT - gpy * DT - Fy[g0] * DT;
}

// ---------------------------------------------------------------------------
extern "C" void kernel_launch(void* const* d_in, const int* in_sizes, int n_in,
                              void* d_out, int out_size, void* d_ws, size_t ws_size,
                              hipStream_t stream) {
    const int n = NN;
    const size_t N2 = (size_t)n * n;
    const float* u0 = (const float*)d_in[0];
    const float* v0 = (const float*)d_in[1];
    const float* p0 = (const float*)d_in[2];
    const float* Fx = (const float*)d_in[3];
    const float* Fy = (const float*)d_in[4];
    const float* k1 = (const float*)d_in[5];
    const float* Wx = (const float*)d_in[6];
    const float* Wy = (const float*)d_in[7];
    const float* Wd = (const float*)d_in[8];
    const float* WA = (const float*)d_in[9];
    const float* Wr = (const float*)d_in[10];
    const int ITER = 5;  // python scalar `iteration`; drives launch count -> host constant

    float* out_u = (float*)d_out;
    float* out_v = out_u + N2;
    float* out_p = out_u + 2 * N2;
    float* out_w = out_u + 3 * N2;
    float* out_r = out_u + 4 * N2;   // 4 floats (2x2 coarsest residual)

    float* ws = (float*)d_ws;
    float* bu = ws;                  // dead after momentum2 -> reused as w ping buffer
    float* bv = ws + N2;             // dead after momentum2 -> reused as p ping buffer
    float* b  = ws + 2 * N2;
    float* rp[11];
    rp[0] = ws + 3 * N2;             // fine residual
    {
        size_t off = 4 * N2;
        for (int l = 1; l <= 10; ++l) {
            rp[l] = ws + off;
            off += (size_t)(n >> l) * (size_t)(n >> l);
        }
    }
    float* wbufA = bu;
    float* ptmp  = bv;

    dim3 b16(16, 16), g128(n / 16, n / 16);
    momentum1_k<<<g128, b16, 0, stream>>>(u0, v0, p0, k1, Wx, Wy, Wd, n, bu, bv);
    momentum2_k<<<g128, b16, 0, stream>>>(u0, v0, p0, bu, bv, k1, Fx, Fy,
                                          Wx, Wy, Wd, n, out_u, out_v);
    divb_k<<<g128, b16, 0, stream>>>(out_u, out_v, Wx, Wy, n, b);

    const float* pcur = p0;
    dim3 gW(n / 32, n / 32);
    for (int it = 0; it < ITER; ++it) {
        residual_wmma_k<<<gW, dim3(128), 0, stream>>>(pcur, b, WA, n, rp[0]);
        for (int l = 1; l <= 10; ++l) {
            int ni = n >> (l - 1), no = ni >> 1;
            dim3 gr((no + 15) / 16, (no + 15) / 16);
            restrict_k<<<gr, b16, 0, stream>>>(rp[l - 1], ni, Wr, rp[l]);
        }
        // V-sweep up: 10 steps, alternating buffers; step 10 lands in out_w
        float* bufs[2] = { wbufA, out_w };
        for (int k = 1; k <= 10; ++k) {
            int i = 11 - k;
            int s = n >> i;                       // input level size: 2,4,...,1024
            const float* wsrc = (k == 1) ? rp[10] : bufs[k & 1];
            float* wdst = bufs[1 - (k & 1)];
            if (s >= 32) {
                smooth_up_wmma_k<<<dim3(s / 32, s / 32), dim3(128), 0, stream>>>(
                    wsrc, rp[i], WA, s, wdst);
            } else {
                smooth_small_k<<<1, dim3(s, s), 0, stream>>>(
                    wsrc, rp[i], WA, s, (k == 1) ? 1 : 0, wdst);
            }
        }
        sub_k<<<(int)((N2 + 255) / 256), 256, 0, stream>>>(pcur, out_w, ptmp, (int)N2);
        jacobi_wmma_k<<<gW, dim3(128), 0, stream>>>(ptmp, b, WA, n, out_p);
        pcur = out_p;
    }
    copy_k<<<1, 32, 0, stream>>>(rp[10], out_r, 4);
    pgrad_correct_k<<<g128, b16, 0, stream>>>(out_p, Wx, Wy, n, out_u, out_v);
}